// InverseHaarWaveletTransform_2774548873638
// MI455X (gfx1250) — compile-verified
//
#include <hip/hip_runtime.h>

// Inverse 2x2 Haar synthesis via V_WMMA_F32_16X16X4_F32.
//
//   [a b c d]^T = S(4x4, +-1) * [ll lh hl hh]^T  per spatial site.
//
// WMMA embedding (D = A(16x4) x B(4x16) + 0):
//   B columns n=0..15  = 16 consecutive w positions of one input row h
//   B rows K0..K3      = ll, lh, hl, hh
//   A rows m           = sign row S[m&3]  (so rows 0-3 and 8-11 both give a,b,c,d)
//   C[m,n]             = component (m&3) at position n
//
// ISA VGPR layouts (CDNA5 ISA):
//   A 16x4 : lanes 0-15 hold K=0,1 in v0,v1 ; lanes 16-31 hold K=2,3
//   B 4x16 : v0 = row K0 (lanes 0-15) / K2 (lanes 16-31); v1 = K1 / K3
//   C 16x16: VGPR r = row r (lanes 0-15) / row r+8 (lanes 16-31)
// => lane n<16  has a,b,c,d in c[0..3]  -> stores {a,b} to out row 2h
//    lane n+16  has a,b,c,d in c[0..3]  -> stores {c,d} to out row 2h+1
// Each half-wave store is a contiguous, 128B-aligned 128-byte segment (NT).

typedef __attribute__((ext_vector_type(2))) float v2f;
typedef __attribute__((ext_vector_type(8))) float v8f;

__global__ __launch_bounds__(256) void ihaar_wmma_kernel(
    const float* __restrict__ ll,   // (planes, 128, 128)
    const float* __restrict__ hf,   // (planes, 3, 128, 128)  [lh, hl, hh]
    float* __restrict__ out,        // (planes, 256, 256)
    int planes) {
  constexpr int H = 128;
  constexpr int W = 128;
  constexpr int PLANE = H * W;

  const int lane = threadIdx.x & 31;
  const int wave = threadIdx.x >> 5;
  const int row_id = blockIdx.x * 8 + wave;   // one (plane, h) row per wave
  const int plane = row_id >> 7;              // row_id / H
  const int h = row_id & (H - 1);
  if (plane >= planes) return;                // wave-uniform; EXEC stays all-ones

  const int n = lane & 15;                    // matrix column = position in tile
  const bool lower = lane < 16;               // lower half-wave?

  // ---- A matrix: per-lane sign constants. comp = row & 3; k order ll,lh,hl,hh
  //   a: + - - +    b: + - + -    c: + + - -    d: + + + +
  const int comp = n & 3;
  v2f Aop;
  if (lower) {
    Aop.x = 1.0f;                                        // k=0 (ll): always +
    Aop.y = (comp < 2) ? -1.0f : 1.0f;                   // k=1 (lh)
  } else {
    Aop.x = (comp & 1) ? 1.0f : -1.0f;                   // k=2 (hl)
    Aop.y = (comp == 0 || comp == 3) ? 1.0f : -1.0f;     // k=3 (hh)
  }

  // ---- per-lane source pointers for the B matrix rows this lane supplies
  const float* llp = ll + (size_t)plane * PLANE + (size_t)h * W;
  const float* hfp = hf + (size_t)plane * (3 * PLANE) + (size_t)h * W;  // lh
  const float* p0 = lower ? llp : (hfp + PLANE);        // K0 = ll | K2 = hl
  const float* p1 = lower ? hfp : (hfp + 2 * PLANE);    // K1 = lh | K3 = hh

  // ---- output pointer: lanes 0-15 -> row 2h, lanes 16-31 -> row 2h+1
  float* outp = out + (size_t)plane * (4 * PLANE)
                    + (size_t)(2 * h + (lower ? 0 : 1)) * (2 * W)
                    + 2 * n;

  // 8 tiles of 16 positions cover the W=128 row
  for (int w0 = 0; w0 < W; w0 += 16) {
    v2f Bop;
    Bop.x = p0[w0 + n];
    Bop.y = p1[w0 + n];

    v8f acc = {};
    // 8 args: (neg_a, A, neg_b, B, c_mod, C, reuse_a, reuse_b)
    acc = __builtin_amdgcn_wmma_f32_16x16x4_f32(
        false, Aop, false, Bop, (short)0, acc, false, false);

    // lanes 0-15: {a,b} = c[0],c[1] ; lanes 16-31: {c,d} = c[2],c[3]
    v2f res;
    res.x = lower ? acc[0] : acc[2];
    res.y = lower ? acc[1] : acc[3];

    // streaming store: full 128B segment per half-wave, bypass caches (NT)
    __builtin_nontemporal_store(res, (v2f*)(outp + 2 * w0));
  }
}

extern "C" void kernel_launch(void* const* d_in, const int* in_sizes, int n_in,
                              void* d_out, int out_size, void* d_ws, size_t ws_size,
                              hipStream_t stream) {
  const float* ll = (const float*)d_in[0];   // (16, 64, 128, 128) f32
  const float* hf = (const float*)d_in[1];   // (16, 192, 128, 128) f32
  float* out = (float*)d_out;                // (16, 64, 256, 256) f32

  const int planes = in_sizes[0] / (128 * 128);   // B*C = 1024
  // one wave per (plane, h) row; 8 waves per block
  const int total_waves = planes * 128;
  dim3 block(256);
  dim3 grid(total_waves / 8);
  ihaar_wmma_kernel<<<grid, block, 0, stream>>>(ll, hf, out, planes);
}